// SSNTXent_46179488367313
// MI455X (gfx1250) — compile-verified
//
#include <hip/hip_runtime.h>
#include <cstdint>

typedef __attribute__((ext_vector_type(2))) float v2f;
typedef __attribute__((ext_vector_type(8))) float v8f;

#define NN 4096
#define DD 256
#define INV_T (1.0f / 0.07f)
#define NCHUNK 8
#define COLS_PER_CHUNK (NN / NCHUNK) /* 512 */

// ---------------------------------------------------------------------------
// Kernel 1: row-normalize features -> fn  (one wave32 per row)
// ---------------------------------------------------------------------------
__global__ __launch_bounds__(256) void k_normalize(const float* __restrict__ feat,
                                                   float* __restrict__ fn) {
  const int lane = threadIdx.x & 31;
  const int row = blockIdx.x * 8 + (threadIdx.x >> 5);
  const float* p = feat + (size_t)row * DD;
  float v[8];
  float s = 0.f;
#pragma unroll
  for (int j = 0; j < 8; ++j) {
    v[j] = p[lane + 32 * j];
    s += v[j] * v[j];
  }
#pragma unroll
  for (int m = 1; m < 32; m <<= 1) s += __shfl_xor(s, m, 32);
  const float scale = 1.0f / fmaxf(sqrtf(s), 1e-8f);
  float* q = fn + (size_t)row * DD;
#pragma unroll
  for (int j = 0; j < 8; ++j) q[lane + 32 * j] = v[j] * scale;
}

// ---------------------------------------------------------------------------
// Kernel 2 (pass 1): neg partials. One wave per (16-row tile, column chunk).
// A tile kept resident in registers; 64 x V_WMMA_F32_16X16X4_F32 per tile.
// ---------------------------------------------------------------------------
__global__ __launch_bounds__(32) void k_pass_neg(const float* __restrict__ fn,
                                                 const float* __restrict__ negm,
                                                 float* __restrict__ negpart) {
  const int lane = threadIdx.x & 31;
  const int lane15 = lane & 15;
  const int hi = lane >> 4;
  const int rowBase = blockIdx.x * 16;
  const int colStart = blockIdx.y * COLS_PER_CHUNK;

  // A fragment preload: lane holds row (rowBase+lane15), K pairs (4k+2*hi, +1)
  const float* ap = fn + (size_t)(rowBase + lane15) * DD + 2 * hi;
  v2f a[64];
#pragma unroll
  for (int kk = 0; kk < 64; ++kk) a[kk] = *(const v2f*)(ap + 4 * kk);

  float accNeg[8];
#pragma unroll
  for (int r = 0; r < 8; ++r) accNeg[r] = 0.f;

  for (int colBase = colStart; colBase < colStart + COLS_PER_CHUNK; colBase += 16) {
    const float* bp = fn + (size_t)(colBase + lane15) * DD + 2 * hi;
    v8f c = {0.f, 0.f, 0.f, 0.f, 0.f, 0.f, 0.f, 0.f};
#pragma unroll
    for (int kk = 0; kk < 64; ++kk) {
      v2f b = *(const v2f*)(bp + 4 * kk);
      c = __builtin_amdgcn_wmma_f32_16x16x4_f32(false, a[kk], false, b,
                                                (short)0, c, false, false);
    }
    const int j = colBase + lane15;
#pragma unroll
    for (int r = 0; r < 8; ++r) {
      const int i = rowBase + 8 * hi + r;  // C layout: VGPR r, half 'hi'
      const float e = __expf(c[r] * INV_T);
      const float m =
          (i == j) ? 0.f : __builtin_nontemporal_load(negm + (size_t)i * NN + j);
      accNeg[r] += e * m;
    }
  }
  // reduce across the 16 lanes sharing each row, then lanes 0/16 write
#pragma unroll
  for (int r = 0; r < 8; ++r) {
    float v = accNeg[r];
    v += __shfl_xor(v, 1, 16);
    v += __shfl_xor(v, 2, 16);
    v += __shfl_xor(v, 4, 16);
    v += __shfl_xor(v, 8, 16);
    if (lane15 == 0)
      negpart[(size_t)blockIdx.y * NN + rowBase + 8 * hi + r] = v;
  }
}

// ---------------------------------------------------------------------------
// Kernel 3: neg[row] = ordered sum of 8 chunk partials (deterministic)
// ---------------------------------------------------------------------------
__global__ __launch_bounds__(256) void k_reduce_neg(const float* __restrict__ negpart,
                                                    float* __restrict__ neg) {
  const int row = blockIdx.x * 256 + threadIdx.x;
  float s = 0.f;
#pragma unroll
  for (int c2 = 0; c2 < NCHUNK; ++c2) s += negpart[(size_t)c2 * NN + row];
  neg[row] = s;
}

// ---------------------------------------------------------------------------
// Kernel 4 (pass 2): log-prob + pos_sum partials (same GEMM recomputed)
// ---------------------------------------------------------------------------
__global__ __launch_bounds__(32) void k_pass_lp(const float* __restrict__ fn,
                                                const float* __restrict__ posm,
                                                const float* __restrict__ neg,
                                                float* __restrict__ lppart,
                                                float* __restrict__ pspart) {
  const int lane = threadIdx.x & 31;
  const int lane15 = lane & 15;
  const int hi = lane >> 4;
  const int rowBase = blockIdx.x * 16;
  const int colStart = blockIdx.y * COLS_PER_CHUNK;

  const float* ap = fn + (size_t)(rowBase + lane15) * DD + 2 * hi;
  v2f a[64];
#pragma unroll
  for (int kk = 0; kk < 64; ++kk) a[kk] = *(const v2f*)(ap + 4 * kk);

  float negrow[8];
#pragma unroll
  for (int r = 0; r < 8; ++r) negrow[r] = neg[rowBase + 8 * hi + r];

  float accLp[8], accPs[8];
#pragma unroll
  for (int r = 0; r < 8; ++r) {
    accLp[r] = 0.f;
    accPs[r] = 0.f;
  }

  for (int colBase = colStart; colBase < colStart + COLS_PER_CHUNK; colBase += 16) {
    const float* bp = fn + (size_t)(colBase + lane15) * DD + 2 * hi;
    v8f c = {0.f, 0.f, 0.f, 0.f, 0.f, 0.f, 0.f, 0.f};
#pragma unroll
    for (int kk = 0; kk < 64; ++kk) {
      v2f b = *(const v2f*)(bp + 4 * kk);
      c = __builtin_amdgcn_wmma_f32_16x16x4_f32(false, a[kk], false, b,
                                                (short)0, c, false, false);
    }
    const int j = colBase + lane15;
#pragma unroll
    for (int r = 0; r < 8; ++r) {
      const int i = rowBase + 8 * hi + r;
      const float e = __expf(c[r] * INV_T);
      const float m =
          (i == j) ? 0.f : __builtin_nontemporal_load(posm + (size_t)i * NN + j);
      const float t = (e * m) / negrow[r] + 1e-20f;
      accLp[r] += m * __logf(t);  // m==0 -> term 0, matches reference
      accPs[r] += m;
    }
  }
#pragma unroll
  for (int r = 0; r < 8; ++r) {
    float vl = accLp[r], vp = accPs[r];
    vl += __shfl_xor(vl, 1, 16); vp += __shfl_xor(vp, 1, 16);
    vl += __shfl_xor(vl, 2, 16); vp += __shfl_xor(vp, 2, 16);
    vl += __shfl_xor(vl, 4, 16); vp += __shfl_xor(vp, 4, 16);
    vl += __shfl_xor(vl, 8, 16); vp += __shfl_xor(vp, 8, 16);
    if (lane15 == 0) {
      const size_t idx = (size_t)blockIdx.y * NN + rowBase + 8 * hi + r;
      lppart[idx] = vl;
      pspart[idx] = vp;
    }
  }
}

// ---------------------------------------------------------------------------
// Kernel 5: final deterministic reduction -> scalar loss
// ---------------------------------------------------------------------------
__global__ __launch_bounds__(256) void k_finalize(const float* __restrict__ lppart,
                                                  const float* __restrict__ pspart,
                                                  float* __restrict__ out) {
  __shared__ float sh[256];
  const int t = threadIdx.x;
  float local = 0.f;
  for (int row = t; row < NN; row += 256) {
    float lp = 0.f, ps = 0.f;
#pragma unroll
    for (int c2 = 0; c2 < NCHUNK; ++c2) {
      lp += lppart[(size_t)c2 * NN + row];
      ps += pspart[(size_t)c2 * NN + row];
    }
    if (ps == 0.f) ps = 1.f;
    local += lp / ps;
  }
  sh[t] = local;
  __syncthreads();
  for (int s = 128; s > 0; s >>= 1) {
    if (t < s) sh[t] += sh[t + s];
    __syncthreads();
  }
  if (t == 0) out[0] = -sh[0] / (float)NN;
}

// ---------------------------------------------------------------------------
extern "C" void kernel_launch(void* const* d_in, const int* in_sizes, int n_in,
                              void* d_out, int out_size, void* d_ws, size_t ws_size,
                              hipStream_t stream) {
  const float* feat = (const float*)d_in[0];  // [N, D]
  const float* posm = (const float*)d_in[1];  // [N, N]
  const float* negm = (const float*)d_in[2];  // [N, N]
  float* out = (float*)d_out;                 // [1]

  float* fn = (float*)d_ws;                      // N*D
  float* negpart = fn + (size_t)NN * DD;         // NCHUNK*N
  float* neg = negpart + (size_t)NCHUNK * NN;    // N
  float* lppart = neg + NN;                      // NCHUNK*N
  float* pspart = lppart + (size_t)NCHUNK * NN;  // NCHUNK*N

  k_normalize<<<NN / 8, 256, 0, stream>>>(feat, fn);

  dim3 grid(NN / 16, NCHUNK);
  k_pass_neg<<<grid, 32, 0, stream>>>(fn, negm, negpart);
  k_reduce_neg<<<NN / 256, 256, 0, stream>>>(negpart, neg);
  k_pass_lp<<<grid, 32, 0, stream>>>(fn, posm, neg, lppart, pspart);
  k_finalize<<<1, 256, 0, stream>>>(lppart, pspart, out);
  (void)in_sizes; (void)n_in; (void)out_size; (void)ws_size;
}